// DoGGradEnhance_38371237822842
// MI455X (gfx1250) — compile-verified
//
#include <hip/hip_runtime.h>
#include <hip/hip_bf16.h>
#include <math.h>

// ---------------------------------------------------------------------------
// DoG-gradient enhancement, MI455X (gfx1250, wave32).
//   K1: gray = mean over 64 channels, via V_WMMA_F32_16X16X4_F32
//   K2: dog (fused 5x5 DoG) + sobel + sigmoid gate -> att plane (LDS-tiled)
//   K3: out = x * (1 + att)   (x re-read hits 192MB L2)
// ---------------------------------------------------------------------------

typedef __attribute__((ext_vector_type(2))) float v2f;
typedef __attribute__((ext_vector_type(8))) float v8f;

#define B_  8
#define C_  64
#define H_  256
#define W_  256
#define HW_ (H_ * W_)            // 65536
#define NPIX (B_ * HW_)          // 524288
#define NELEM ((size_t)B_ * C_ * HW_)  // 33554432

// ---------------------------------------------------------------------------
// K1: channel mean via fp32 WMMA.
// Each wave handles 16 consecutive pixels (one W-row segment, same batch).
// A-matrix (16x4 f32) layout per CDNA5 ISA: lanes 0-15 -> M=0..15, VGPR0=K0,
// VGPR1=K1; lanes 16-31 -> same M, K2/K3.  B = all 1/64 (layout-invariant).
// 16 accumulating WMMAs cover 64 channels.  D[m][n] = mean(pixel m) for all n;
// lane 0 holds D[0..7][0] in acc[0..7], lane 16 holds D[8..15][0].
// ---------------------------------------------------------------------------
__global__ void dog_gray_mean_wmma(const float* __restrict__ x,
                                   float* __restrict__ gray) {
  const int tid  = blockIdx.x * blockDim.x + threadIdx.x;
  const int wid  = tid >> 5;          // wave id: 32768 waves total
  const int lane = tid & 31;

  const int p0 = wid << 4;            // base pixel in flat (B*H*W) space
  const int b  = p0 >> 16;            // batch (HW_ = 65536 pixels/batch)
  const int s  = p0 & (HW_ - 1);      // spatial offset in batch

  const int m     = lane & 15;        // pixel within tile
  const int khalf = lane >> 4;        // 0 -> K{0,1}, 1 -> K{2,3}

  const float* xb = x + (size_t)b * C_ * HW_ + s + m;   // channel 0, this pixel

  v2f bmat;  bmat.x = 1.0f / 64.0f;  bmat.y = 1.0f / 64.0f;
  v8f acc = {};

#pragma unroll
  for (int cg = 0; cg < 16; ++cg) {
    const int c0 = (cg << 2) + (khalf << 1);
    v2f a;
    a.x = xb[(size_t)c0 * HW_];
    a.y = xb[(size_t)(c0 + 1) * HW_];
    acc = __builtin_amdgcn_wmma_f32_16x16x4_f32(
        /*neg_a=*/false, a, /*neg_b=*/false, bmat,
        /*c_mod=*/(short)0, acc, /*reuse_a=*/false, /*reuse_b=*/false);
  }

  // Column N==0 of D lives in lanes 0 (M=0..7) and 16 (M=8..15).
  if ((lane & 15) == 0) {
    float* g = gray + p0 + (khalf << 3);
    float4 lo = make_float4(acc[0], acc[1], acc[2], acc[3]);
    float4 hi = make_float4(acc[4], acc[5], acc[6], acc[7]);
    *(float4*)(g)     = lo;
    *(float4*)(g + 4) = hi;
  }
}

// ---------------------------------------------------------------------------
// K2: att plane.  32x32 output tile per 256-thread block.
//   sgray: 38x38 halo (dog needs +-1, gray needs +-2 around that)
//   sdog : 34x34 (zero outside image => matches reference's staged zero-pad)
// ---------------------------------------------------------------------------
__global__ void dog_att_kernel(const float* __restrict__ gray,
                               const float* __restrict__ gate_w,
                               const float* __restrict__ gate_b,
                               float* __restrict__ att) {
  __shared__ float sgray[38 * 40];
  __shared__ float sdog[34 * 36];
  __shared__ float skd[25];

  const int tid = threadIdx.x;
  const int tx0 = blockIdx.x << 5;
  const int ty0 = blockIdx.y << 5;
  const int b   = blockIdx.z;
  const float* gb = gray + (size_t)b * HW_;

  // Fused DoG 5x5 kernel weights (gk1 - gk2), computed once per block.
  if (tid < 25) {
    const int i = tid / 5, j = tid % 5;
    float s1 = 0.f, s2 = 0.f;
    for (int ii = 0; ii < 5; ++ii)
      for (int jj = 0; jj < 5; ++jj) {
        const float dd = (float)((ii - 2) * (ii - 2) + (jj - 2) * (jj - 2));
        s1 += __expf(-dd * 0.5f);    // sigma1 = 1
        s2 += __expf(-dd * 0.125f);  // sigma2 = 2
      }
    const float d2 = (float)((i - 2) * (i - 2) + (j - 2) * (j - 2));
    skd[tid] = __expf(-d2 * 0.5f) / s1 - __expf(-d2 * 0.125f) / s2;
  }

  // Stage gray halo (zero-padded outside image).
  for (int idx = tid; idx < 38 * 38; idx += 256) {
    const int r = idx / 38, c = idx % 38;
    const int gy = ty0 - 3 + r, gx = tx0 - 3 + c;
    float v = 0.f;
    if (gy >= 0 && gy < H_ && gx >= 0 && gx < W_) v = gb[gy * W_ + gx];
    sgray[r * 40 + c] = v;
  }
  __syncthreads();

  // dog tile: rows/cols (ty0-1 .. ty0+32) x (tx0-1 .. tx0+32); 0 outside image.
  for (int idx = tid; idx < 34 * 34; idx += 256) {
    const int r = idx / 34, c = idx % 34;
    const int gy = ty0 - 1 + r, gx = tx0 - 1 + c;
    float v = 0.f;
    if (gy >= 0 && gy < H_ && gx >= 0 && gx < W_) {
#pragma unroll
      for (int ki = 0; ki < 5; ++ki)
#pragma unroll
        for (int kj = 0; kj < 5; ++kj)
          v = fmaf(skd[ki * 5 + kj], sgray[(r + ki) * 40 + (c + kj)], v);
    }
    sdog[r * 36 + c] = v;
  }
  __syncthreads();

  const float gw  = gate_w[0];
  const float gbv = gate_b[0];

  for (int idx = tid; idx < 32 * 32; idx += 256) {
    const int i = idx >> 5, j = idx & 31;   // local output coords
    const float d00 = sdog[i * 36 + j];
    const float d01 = sdog[i * 36 + j + 1];
    const float d02 = sdog[i * 36 + j + 2];
    const float d10 = sdog[(i + 1) * 36 + j];
    const float d12 = sdog[(i + 1) * 36 + j + 2];
    const float d20 = sdog[(i + 2) * 36 + j];
    const float d21 = sdog[(i + 2) * 36 + j + 1];
    const float d22 = sdog[(i + 2) * 36 + j + 2];
    const float gxv = (d02 - d00) + 2.f * (d12 - d10) + (d22 - d20);
    const float gyv = (d20 - d00) + 2.f * (d21 - d01) + (d22 - d02);
    const float mag = sqrtf(gxv * gxv + gyv * gyv + 1e-6f);
    const float a = 1.f / (1.f + __expf(-(mag * gw + gbv)));
    att[(size_t)b * HW_ + (ty0 + i) * W_ + (tx0 + j)] = a;
  }
}

// ---------------------------------------------------------------------------
// K3: out = x * (1 + att), broadcast over 64 channels. float4-vectorized.
// x re-read should be served from the 192MB L2 (loaded by K1).
// ---------------------------------------------------------------------------
__global__ void dog_modulate_kernel(const float* __restrict__ x,
                                    const float* __restrict__ att,
                                    float* __restrict__ out) {
  const size_t t  = (size_t)blockIdx.x * blockDim.x + threadIdx.x;
  const size_t i4 = t << 2;                       // 4 floats per thread
  const int b = (int)(i4 >> 22);                  // C_*HW_ = 4194304 / batch
  const int s = (int)(i4 & (size_t)(HW_ - 1));    // spatial (multiple of 4)

  __builtin_prefetch(x + i4 + 16384, 0, 1);       // global_prefetch_b8

  const float4 xv = *(const float4*)(x + i4);
  const float4 av = *(const float4*)(att + ((size_t)b << 16) + s);
  float4 ov;
  ov.x = xv.x * (1.f + av.x);
  ov.y = xv.y * (1.f + av.y);
  ov.z = xv.z * (1.f + av.z);
  ov.w = xv.w * (1.f + av.w);
  *(float4*)(out + i4) = ov;
}

// ---------------------------------------------------------------------------
extern "C" void kernel_launch(void* const* d_in, const int* in_sizes, int n_in,
                              void* d_out, int out_size, void* d_ws, size_t ws_size,
                              hipStream_t stream) {
  const float* x      = (const float*)d_in[0];
  const float* gate_w = (const float*)d_in[1];
  const float* gate_b = (const float*)d_in[2];
  float* out = (float*)d_out;

  float* gray = (float*)d_ws;           // NPIX floats (2 MB)
  float* att  = gray + NPIX;            // NPIX floats (2 MB)

  // K1: 32768 waves * 16 px = 524288 px; 256 thr/block -> 4096 blocks.
  dog_gray_mean_wmma<<<NPIX / 128, 256, 0, stream>>>(x, gray);

  // K2: 8x8 tiles per image, 8 batches.
  dim3 g2(W_ / 32, H_ / 32, B_);
  dog_att_kernel<<<g2, 256, 0, stream>>>(gray, gate_w, gate_b, att);

  // K3: 33554432 / 4 = 8388608 threads -> 32768 blocks.
  dog_modulate_kernel<<<(int)(NELEM / 1024), 256, 0, stream>>>(x, att, out);
}